// simpleGCN_19318762897564
// MI455X (gfx1250) — compile-verified
//
#include <hip/hip_runtime.h>
#include <hip/hip_bf16.h>

typedef __attribute__((ext_vector_type(2))) float v2f;
typedef __attribute__((ext_vector_type(8))) float v8f;

#define N_NODES 50000
#define NUM_FEATS 512
#define NUM_CLASSES 64

#define CHUNK_K 64   // K rows of W staged per chunk
#define LDS_ROW 72   // float stride per W row in LDS (64 + 8 pad: bank-conflict-free)

// ---------------------------------------------------------------------------
// Utility: grid-stride zero fill
// ---------------------------------------------------------------------------
__global__ void zero_f32(float* __restrict__ p, int n) {
  int i = blockIdx.x * blockDim.x + threadIdx.x;
  int stride = gridDim.x * blockDim.x;
  for (; i < n; i += stride) p[i] = 0.0f;
}

// ---------------------------------------------------------------------------
// Degree accumulation: deg[row[e]] += 1
// ---------------------------------------------------------------------------
__global__ void degree_kernel(const long long* __restrict__ row,
                              float* __restrict__ deg, int E) {
  int e = blockIdx.x * blockDim.x + threadIdx.x;
  if (e < E) {
    atomicAdd(&deg[(int)row[e]], 1.0f);
  }
}

// deg -> deg^{-1/2} in place (0 where deg==0)
__global__ void dinv_kernel(float* __restrict__ deg, int n) {
  int i = blockIdx.x * blockDim.x + threadIdx.x;
  if (i < n) {
    float d = deg[i];
    deg[i] = (d > 0.0f) ? rsqrtf(fmaxf(d, 1e-12f)) : 0.0f;
  }
}

// norm[e] = dinv[row[e]] * dinv[col[e]]
__global__ void norm_kernel(const long long* __restrict__ row,
                            const long long* __restrict__ col,
                            const float* __restrict__ dinv,
                            float* __restrict__ norm, int E) {
  int e = blockIdx.x * blockDim.x + threadIdx.x;
  if (e < E) {
    norm[e] = dinv[(int)row[e]] * dinv[(int)col[e]];
  }
}

// ---------------------------------------------------------------------------
// Dense projection H = X[N,512] @ W[512,64] with V_WMMA_F32_16X16X4_F32.
//
// 8 waves / block, one 16-row M tile per wave, four 16x16 accumulators span
// the 64 output columns. W is streamed into LDS with the CDNA5 async-copy
// path (global_load_async_to_lds_b128, ASYNCcnt), double-buffered in 64-K-row
// chunks; B fragments come from LDS (ds_load), X streams straight from HBM.
//
// ISA f32 operand layouts (CDNA5 ISA §7.12.2):
//   A (16x4):  lanes 0-15 -> M=lane, K pair {0,1}; lanes 16-31 -> M=lane-16,
//              K pair {2,3}; VGPR0 = even K, VGPR1 = odd K of the pair.
//   B (4x16):  mirrored: lane halves select K pair, lane&15 = N column.
//   C/D (16x16): VGPR i holds M = i + 8*(lane>=16), N = lane&15.
// ---------------------------------------------------------------------------
__global__ __launch_bounds__(256) void gemm_wmma_f32(
    const float* __restrict__ X, const float* __restrict__ W,
    float* __restrict__ H, int nrows) {
  __shared__ float sW[2][CHUNK_K * LDS_ROW];  // 2 x 18 KB

  const int tid  = threadIdx.x;
  const int lane = tid & 31;
  const int wid  = tid >> 5;
  const int wave = blockIdx.x * 8 + wid;
  const int m0   = wave * 16;
  const bool active = (m0 < nrows);  // tail waves still run barriers/prefetch

  // Async prefetch of one W chunk: CHUNK_K x 64 f32 = 16 KB = 1024 x b128.
  // 256 threads x 4 transfers; LDS rows padded to LDS_ROW floats.
  auto prefetch = [&](int buf, int kbase) {
#pragma unroll
    for (int i = 0; i < 4; ++i) {
      int t = tid + i * 256;
      int r = t >> 4;       // W row within chunk (16 b128 per 64-float row)
      int s = t & 15;       // 16-byte segment within row
      const float* g = W + (size_t)(kbase + r) * NUM_CLASSES + s * 4;
      unsigned int ldsoff =
          (unsigned int)(uintptr_t)(&sW[buf][r * LDS_ROW + s * 4]);
      asm volatile("global_load_async_to_lds_b128 %0, %1, off"
                   :: "v"(ldsoff), "v"(g)
                   : "memory");
    }
  };

  const int mrow = lane & 15;         // M row this lane supplies for A
  const int ncol = lane & 15;         // N column for B / C
  const int kb   = (lane >> 4) << 1;  // 0 or 2: K pair within a k-step

  const int mclamp = active ? m0 : 0;
  const float* xrow = X + (size_t)(mclamp + mrow) * NUM_FEATS;

  v8f acc0 = {}, acc1 = {}, acc2 = {}, acc3 = {};

  prefetch(0, 0);
  asm volatile("s_wait_asynccnt 0" ::: "memory");
  __syncthreads();

  const int nchunks = NUM_FEATS / CHUNK_K;  // 8
  for (int c = 0; c < nchunks; ++c) {
    const int cur = c & 1;
    if (c + 1 < nchunks) prefetch(cur ^ 1, (c + 1) * CHUNK_K);

    if (active) {
      const float* xc = xrow + c * CHUNK_K;
#pragma unroll 4
      for (int kk = 0; kk < CHUNK_K; kk += 4) {
        v2f a;
        a.x = xc[kk + kb];
        a.y = xc[kk + kb + 1];

        const float* w0 = &sW[cur][(kk + kb) * LDS_ROW];
        const float* w1 = w0 + LDS_ROW;
        v2f b0 = {w0[ncol],      w1[ncol]};
        v2f b1 = {w0[ncol + 16], w1[ncol + 16]};
        v2f b2 = {w0[ncol + 32], w1[ncol + 32]};
        v2f b3 = {w0[ncol + 48], w1[ncol + 48]};

        acc0 = __builtin_amdgcn_wmma_f32_16x16x4_f32(false, a, false, b0,
                                                     (short)0, acc0, false, false);
        acc1 = __builtin_amdgcn_wmma_f32_16x16x4_f32(false, a, false, b1,
                                                     (short)0, acc1, false, false);
        acc2 = __builtin_amdgcn_wmma_f32_16x16x4_f32(false, a, false, b2,
                                                     (short)0, acc2, false, false);
        acc3 = __builtin_amdgcn_wmma_f32_16x16x4_f32(false, a, false, b3,
                                                     (short)0, acc3, false, false);
      }
    }

    // next chunk landed + everyone done reading current before it is reused
    asm volatile("s_wait_asynccnt 0" ::: "memory");
    __syncthreads();
  }

  if (active) {
    // C/D layout: VGPR i -> row (mb + i), col = tile*16 + ncol
    const int mb = (lane >> 4) * 8;
    float* hout = H + (size_t)(m0 + mb) * NUM_CLASSES;
    v8f accs[4] = {acc0, acc1, acc2, acc3};
#pragma unroll
    for (int t = 0; t < 4; ++t) {
#pragma unroll
      for (int i = 0; i < 8; ++i) {
        hout[(size_t)i * NUM_CLASSES + t * 16 + ncol] = accs[t][i];
      }
    }
  }
}

// ---------------------------------------------------------------------------
// One aggregation layer: dst[col[e]] += norm[e] * src[row[e]]
// One wave32 per edge; each lane handles feature f=lane and f=lane+32.
// Gathers are 2x contiguous 128B per wave; scatters are L2 f32 atomics
// (src/dst are 12.8 MB each -> fully L2-resident on the 192 MB L2).
// ---------------------------------------------------------------------------
__global__ __launch_bounds__(256) void spmm_scatter(
    const float* __restrict__ src, float* __restrict__ dst,
    const long long* __restrict__ row, const long long* __restrict__ col,
    const float* __restrict__ norm, int E) {
  const int edge = blockIdx.x * (blockDim.x >> 5) + (threadIdx.x >> 5);
  const int lane = threadIdx.x & 31;
  if (edge >= E) return;

  const int r = (int)row[edge];
  const int c = (int)col[edge];
  const float nv = norm[edge];

  const float* s = src + (size_t)r * NUM_CLASSES;
  float* d = dst + (size_t)c * NUM_CLASSES;

  float v0 = s[lane]      * nv;
  float v1 = s[lane + 32] * nv;
  atomicAdd(&d[lane],      v0);
  atomicAdd(&d[lane + 32], v1);
}

// ---------------------------------------------------------------------------
// Host-side launch
// ---------------------------------------------------------------------------
extern "C" void kernel_launch(void* const* d_in, const int* in_sizes, int n_in,
                              void* d_out, int out_size, void* d_ws, size_t ws_size,
                              hipStream_t stream) {
  const float* x = (const float*)d_in[0];           // [N, 512]
  const float* w = (const float*)d_in[1];           // [512, 64]
  const long long* ei = (const long long*)d_in[2];  // [2, E] int64
  float* out = (float*)d_out;                       // [N, 64]

  const int N = N_NODES;
  const int E = in_sizes[2] / 2;
  const long long* row = ei;
  const long long* col = ei + E;

  // Workspace layout (floats): deg/dinv [N] | norm [E] | hbuf [N*64]
  float* deg  = (float*)d_ws;
  float* norm = deg + N;
  float* hbuf = norm + E;

  // 1) normalization coefficients
  zero_f32<<<256, 256, 0, stream>>>(deg, N);
  degree_kernel<<<(E + 255) / 256, 256, 0, stream>>>(row, deg, E);
  dinv_kernel<<<(N + 255) / 256, 256, 0, stream>>>(deg, N);
  norm_kernel<<<(E + 255) / 256, 256, 0, stream>>>(row, col, deg, norm, E);

  // 2) dense projection via f32 WMMA: 50000/16 = 3125 M tiles, 8 waves/block
  {
    int mtiles = (N + 15) / 16;
    int wavesPerBlock = 8;  // 256 threads
    int blocks = (mtiles + wavesPerBlock - 1) / wavesPerBlock;
    gemm_wmma_f32<<<blocks, 256, 0, stream>>>(x, w, out, N);
  }

  // 3) four aggregation layers, ping-pong d_out <-> hbuf; final lands in d_out
  const int hElems = N * NUM_CLASSES;
  float* src = out;
  float* dst = hbuf;
  for (int l = 0; l < 4; ++l) {
    zero_f32<<<512, 256, 0, stream>>>(dst, hElems);
    int wavesPerBlock = 8;  // 256 threads
    int blocks = (E + wavesPerBlock - 1) / wavesPerBlock;
    spmm_scatter<<<blocks, 256, 0, stream>>>(src, dst, row, col, norm, E);
    float* t = src; src = dst; dst = t;
  }
  // layers: out->hbuf, hbuf->out, out->hbuf, hbuf->out  => result in d_out
}